// CustomCrossEntropyLossTopK_43233140802038
// MI455X (gfx1250) — compile-verified
//
#include <hip/hip_runtime.h>

// ---------------------------------------------------------------------------
// CustomCrossEntropyLossTopK for MI455X (gfx1250, wave32)
// B=16384 rows, C=8192 cols, fp32.  Memory-bound: 512MB @ 23.3TB/s ~= 22us.
// One wave per row; streaming b128 NT loads; online logsumexp + top-5;
// deterministic two-stage reduction (WMMA f32 16x16x4 exact sum at the end).
// ---------------------------------------------------------------------------

typedef __attribute__((ext_vector_type(4))) float v4f;
typedef __attribute__((ext_vector_type(2))) float v2f;
typedef __attribute__((ext_vector_type(8))) float v8f;
typedef int v4i_vs __attribute__((vector_size(16)));   // matches builtin param

#define B_ROWS 16384
#define C_COLS 8192
#define TOPK_N 5

__device__ __forceinline__ float shflx_f(float v, int m) { return __shfl_xor(v, m, 32); }
__device__ __forceinline__ int   shflx_i(int v, int m)   { return __shfl_xor(v, m, 32); }

__global__ __launch_bounds__(256) void rowstats_kernel(
    const float* __restrict__ input, const int* __restrict__ target,
    float* __restrict__ row_terms)
{
    const int wave = threadIdx.x >> 5;
    const int lane = threadIdx.x & 31;
    const int row  = blockIdx.x * 8 + wave;
    const float* __restrict__ rp = input + (size_t)row * C_COLS;

    // --- streaming pass: online logsumexp + per-lane sorted top-5 ----------
    float m = -__builtin_inff();
    float s = 0.0f;
    float tv0 = -__builtin_inff(), tv1 = tv0, tv2 = tv0, tv3 = tv0, tv4 = tv0;
    int   ti0 = -1, ti1 = -1, ti2 = -1, ti3 = -1, ti4 = -1;

    #pragma unroll 4
    for (int t = 0; t < C_COLS / 128; ++t) {
        const int col = t * 128 + lane * 4;
        v4f x = __builtin_nontemporal_load((const v4f*)(rp + col));
        #pragma unroll
        for (int j = 0; j < 4; ++j) {
            const float xv = x[j];
            // online logsumexp: 1 exp per element on the common path
            if (xv > m) { s = s * __expf(m - xv) + 1.0f; m = xv; }
            else        { s += __expf(xv - m); }
            // sorted-descending top-5 insertion (rare on random data)
            if (xv > tv4) {
                const int ci = col + j;
                if (xv > tv0)      { tv4=tv3;ti4=ti3; tv3=tv2;ti3=ti2; tv2=tv1;ti2=ti1; tv1=tv0;ti1=ti0; tv0=xv;ti0=ci; }
                else if (xv > tv1) { tv4=tv3;ti4=ti3; tv3=tv2;ti3=ti2; tv2=tv1;ti2=ti1; tv1=xv;ti1=ci; }
                else if (xv > tv2) { tv4=tv3;ti4=ti3; tv3=tv2;ti3=ti2; tv2=xv;ti2=ci; }
                else if (xv > tv3) { tv4=tv3;ti4=ti3; tv3=xv;ti3=ci; }
                else               { tv4=xv;ti4=ci; }
            }
        }
    }

    // --- wave combine of (m, s) --------------------------------------------
    #pragma unroll
    for (int off = 16; off; off >>= 1) {
        const float om = shflx_f(m, off);
        const float os = shflx_f(s, off);
        const float nm = fmaxf(m, om);
        s = s * __expf(m - nm) + os * __expf(om - nm);
        m = nm;
    }
    const float lse = m + logf(s);

    // --- wave top-5 merge: 5 rounds of argmax over per-lane sorted heads ---
    float wv[TOPK_N];
    int   wi[TOPK_N];
    #pragma unroll
    for (int k = 0; k < TOPK_N; ++k) {
        float cv = tv0;
        int   ci = ti0;
        #pragma unroll
        for (int off = 16; off; off >>= 1) {
            const float ov = shflx_f(cv, off);
            const int   oi = shflx_i(ci, off);
            // tie-break on smaller column index (matches top_k ordering)
            if ((ov > cv) || (ov == cv && (unsigned)oi < (unsigned)ci)) { cv = ov; ci = oi; }
        }
        wv[k] = cv; wi[k] = ci;
        if (ci == ti0 && cv == tv0) {   // this lane won: advance its head
            tv0=tv1;ti0=ti1; tv1=tv2;ti1=ti2; tv2=tv3;ti2=ti3; tv3=tv4;ti3=ti4;
            tv4 = -__builtin_inff(); ti4 = -1;
        }
    }

    // --- per-row loss terms -------------------------------------------------
    const int   tgt  = target[row];
    const float xt   = rp[tgt];          // uniform broadcast load (cached)
    const float lp_t = xt - lse;         // log_softmax gathered at target

    const float LN_P = 0.00995033085316808285f;   // ln(1.01)
    const float a0   = wv[0] * LN_P;              // wv sorted desc -> a0 is max
    float es = 0.0f;
    #pragma unroll
    for (int k = 0; k < TOPK_N; ++k) es += __expf(wv[k] * LN_P - a0);
    const float denom = a0 + logf(es);

    float l2 = -lp_t;                    // default: target not in top-5
    #pragma unroll
    for (int k = 0; k < TOPK_N; ++k)
        if (wi[k] == tgt) l2 = -(wv[k] * LN_P - denom);

    if (lane == 0) row_terms[row] = (-lp_t) + l2;
}

// ---------------------------------------------------------------------------
// Deterministic final reduction: 16384 -> 256 (fixed order) -> LDS tree -> 64
// -> exact fp32 sum via V_WMMA_F32_16X16X4_F32 (D = A x ones + 0).
// ---------------------------------------------------------------------------
__global__ __launch_bounds__(256) void reduce_kernel(
    const float* __restrict__ terms, float* __restrict__ out)
{
    __shared__ float red[256];
    const int tid = threadIdx.x;
    float acc = 0.0f;
    #pragma unroll 8
    for (int k = 0; k < B_ROWS / 256; ++k) acc += terms[tid + 256 * k];
    red[tid] = acc;
    __syncthreads();
    if (tid < 128) red[tid] += red[tid + 128];
    __syncthreads();
    if (tid < 64)  red[tid] += red[tid + 64];
    __syncthreads();
    if (tid < 32) {   // one full wave32, EXEC all-ones (WMMA requirement)
#if __has_builtin(__builtin_amdgcn_wmma_f32_16x16x4_f32)
        v2f a; a.x = red[2 * tid]; a.y = red[2 * tid + 1];   // 64 partials -> A(16x4)
        v2f b; b.x = 1.0f; b.y = 1.0f;                        // B = ones(4x16)
        v8f c = {};
        c = __builtin_amdgcn_wmma_f32_16x16x4_f32(
                /*neg_a=*/false, a, /*neg_b=*/false, b,
                /*c_mod=*/(short)0, c, /*reuse_a=*/false, /*reuse_b=*/false);
        // D[i][j] = rowsum_A(i); column j=0 lives in lane 0 (M=0..7) and
        // lane 16 (M=8..15): total = sum of those 16 entries.
        float q = c[0]+c[1]+c[2]+c[3]+c[4]+c[5]+c[6]+c[7];
        q += shflx_f(q, 16);
        if (tid == 0) out[0] = q * (1.0f / (float)B_ROWS);
#else
        float v = red[2 * tid] + red[2 * tid + 1];
        #pragma unroll
        for (int off = 16; off; off >>= 1) v += shflx_f(v, off);
        if (tid == 0) out[0] = v * (1.0f / (float)B_ROWS);
#endif
    }
}

// ---------------------------------------------------------------------------
// CDNA5 async-to-LDS probe (not launched): verifies the async engine path.
// Diagnostic from last round: param0 is `int __vector_size__(16) AS(1)*`.
// ---------------------------------------------------------------------------
#if __has_builtin(__builtin_amdgcn_global_load_async_to_lds_b128)
__global__ void cdna5_async_probe(const float* __restrict__ g, float* __restrict__ o)
{
    __shared__ float buf[1024];
    const float* gp = g + threadIdx.x * 4;
    float*       lp = &buf[threadIdx.x * 4];
    auto g1 = (__attribute__((address_space(1))) v4i_vs*)gp;
    auto l3 = (__attribute__((address_space(3))) v4i_vs*)lp;
    __builtin_amdgcn_global_load_async_to_lds_b128(g1, l3, 0, 0);
#if __has_builtin(__builtin_amdgcn_s_wait_asynccnt)
    __builtin_amdgcn_s_wait_asynccnt(0);
#else
    asm volatile("s_wait_asynccnt 0" ::: "memory");
#endif
    __syncthreads();
    o[threadIdx.x] = buf[threadIdx.x];
}
#endif

// ---------------------------------------------------------------------------
extern "C" void kernel_launch(void* const* d_in, const int* in_sizes, int n_in,
                              void* d_out, int out_size, void* d_ws, size_t ws_size,
                              hipStream_t stream)
{
    const float* input  = (const float*)d_in[0];   // [16384, 8192] fp32
    const int*   target = (const int*)d_in[1];     // [16384] int32
    float*       out    = (float*)d_out;           // scalar fp32
    float*       terms  = (float*)d_ws;            // 16384 fp32 partials

    rowstats_kernel<<<B_ROWS / 8, 256, 0, stream>>>(input, target, terms);
    reduce_kernel<<<1, 256, 0, stream>>>(terms, out);
}